// Spectral_Degradation_17334488007354
// MI455X (gfx1250) — compile-verified
//
#include <hip/hip_runtime.h>
#include <math.h>

// ---------------------------------------------------------------------------
// Spectral_Degradation for MI455X (gfx1250), f32 end-to-end via
// V_WMMA_F32_16X16X4_F32. n=8, C=128, h=w=128, c=8 bands, mid=64, pos_dim=32.
//
// Pass structure (all GEMMs are WMMA 16x16 tiles, K stepped by 4):
//   k_emb          : tanh positional embedding -> ws
//   k_modulate     : xmod = clip(gelu(Wm2 @ gelu(Wm1 @ [x;emb])))      (ws)
//   k_band_stats   : h1 = W1[b] @ xmod, per-(n,b,m) sum/sumsq partials
//   k_reduce_stats : partials -> mean, rsqrt(var+eps)
//   k_bands        : h1 -> instance-norm -> gelu -> W2 -> gelu -> W3 -> out
//
// Band kernels: one block = (image, 128-pixel chunk); wave w = band w, so the
// 8 waves issue identical xmod (B-operand) loads -> WGP-cache hits; xmod is
// streamed from HBM/L2 once per pass (roofline minimum).
// ---------------------------------------------------------------------------

typedef __attribute__((ext_vector_type(2))) float v2f;
typedef __attribute__((ext_vector_type(8))) float v8f;

#define HW      16384      // 128*128
#define NBATCH  8
#define CCH     128
#define MID     64
#define PDIM    32
#define NBAND   8

__device__ __forceinline__ v8f wmma4(v2f a, v2f b, v8f c) {
  // 8-arg f32-family pattern: (neg_a, A, neg_b, B, c_mod, C, reuse_a, reuse_b)
  return __builtin_amdgcn_wmma_f32_16x16x4_f32(false, a, false, b,
                                               (short)0, c, false, false);
}

__device__ __forceinline__ float gelu_f(float x) {
  return 0.5f * x * (1.0f + erff(x * 0.70710678118654752440f));
}
__device__ __forceinline__ float clamp01(float x) {
  return fminf(fmaxf(x, 0.0f), 1.0f);
}

// ---------------------------------------------------------------------------
// 1) positional embedding: emb[pd, p] = tanh(Wp[pd,0]*ym + Wp[pd,1]*xm)
// ---------------------------------------------------------------------------
__global__ void k_emb(const float* __restrict__ Wp, float* __restrict__ emb) {
  int p = blockIdx.x * blockDim.x + threadIdx.x;
  if (p >= HW) return;
  int i = p >> 7, j = p & 127;
  float ym = -1.0f + (2.0f / 127.0f) * (float)i;
  float xm = -1.0f + (2.0f / 127.0f) * (float)j;
#pragma unroll 4
  for (int pd = 0; pd < PDIM; ++pd) {
    float v = tanhf(Wp[pd * 2 + 0] * ym + Wp[pd * 2 + 1] * xm);
    emb[(size_t)pd * HW + p] = v;
  }
}

// ---------------------------------------------------------------------------
// 2) modulation: xmod = clip(gelu(Wm2 @ gelu(Wm1 @ [x;emb])), 0, 1)
//    One wave owns a 16-pixel column. K loop split at the x/emb boundary so
//    both loops are exec-uniform (no divergent branches around WMMA).
// ---------------------------------------------------------------------------
__global__ __launch_bounds__(256) void k_modulate(
    const float* __restrict__ x, const float* __restrict__ emb,
    const float* __restrict__ Wm1, const float* __restrict__ Wm2,
    float* __restrict__ xmod) {
  __shared__ float lds[8][MID * 17];
  const int wave = threadIdx.x >> 5;
  const int lane = threadIdx.x & 31;
  const int n    = blockIdx.x >> 7;                 // 128 blocks / image
  const int pb   = (blockIdx.x & 127) * 128 + wave * 16;
  const int lm   = lane & 15;
  const int kh   = (lane >> 4) * 2;                 // K sub-offset per half
  const int mh   = (lane >> 4) * 8;                 // M offset in C/D tiles
  const int pix  = pb + lm;
  const float* xn = x + (size_t)n * CCH * HW;

  // GEMM1: [64 x 160] x [160 x 16]  (K = 0..127 from x, 128..159 from emb)
  v8f acc[4] = {};
  for (int k0 = 0; k0 < CCH; k0 += 4) {
    int kk = k0 + kh;
    v2f b;
    b.x = xn[(size_t)kk * HW + pix];
    b.y = xn[(size_t)(kk + 1) * HW + pix];
#pragma unroll
    for (int t = 0; t < 4; ++t) {
      v2f a = *(const v2f*)(Wm1 + (t * 16 + lm) * (CCH + PDIM) + kk);
      acc[t] = wmma4(a, b, acc[t]);
    }
  }
  for (int k0 = 0; k0 < PDIM; k0 += 4) {
    int kk = k0 + kh;
    v2f b;
    b.x = emb[(size_t)kk * HW + pix];
    b.y = emb[(size_t)(kk + 1) * HW + pix];
#pragma unroll
    for (int t = 0; t < 4; ++t) {
      v2f a = *(const v2f*)(Wm1 + (t * 16 + lm) * (CCH + PDIM) + CCH + kk);
      acc[t] = wmma4(a, b, acc[t]);
    }
  }
  float* sm = &lds[wave][0];
#pragma unroll
  for (int t = 0; t < 4; ++t)
#pragma unroll
    for (int v = 0; v < 8; ++v)
      sm[(t * 16 + v + mh) * 17 + lm] = gelu_f(acc[t][v]);

  // GEMM2: [128 x 64] x [64 x 16], B from per-wave LDS (same-wave in-order)
  v8f acc2[8] = {};
  for (int k0 = 0; k0 < MID; k0 += 4) {
    int kk = k0 + kh;
    v2f b;
    b.x = sm[kk * 17 + lm];
    b.y = sm[(kk + 1) * 17 + lm];
#pragma unroll
    for (int t = 0; t < 8; ++t) {
      v2f a = *(const v2f*)(Wm2 + (t * 16 + lm) * MID + kk);
      acc2[t] = wmma4(a, b, acc2[t]);
    }
  }
  float* xo = xmod + (size_t)n * CCH * HW;
#pragma unroll
  for (int t = 0; t < 8; ++t)
#pragma unroll
    for (int v = 0; v < 8; ++v) {
      int m = t * 16 + v + mh;
      xo[(size_t)m * HW + pix] = clamp01(gelu_f(acc2[t][v]));
    }
}

// ---------------------------------------------------------------------------
// 3) band conv1 stats pass. Block = (image, 128-pixel chunk); wave = band.
//    Running sum/sumsq held as v8f across the 8 sub-tiles, one shuffle-
//    reduce at the end; deterministic per-chunk partials, no atomics.
// ---------------------------------------------------------------------------
__global__ __launch_bounds__(256) void k_band_stats(
    const float* __restrict__ xmod, const float* __restrict__ W1,
    float* __restrict__ psum, float* __restrict__ psq) {
  const int band  = threadIdx.x >> 5;               // wave = band
  const int lane  = threadIdx.x & 31;
  const int n     = blockIdx.x >> 7;
  const int chunk = blockIdx.x & 127;               // 128-pixel chunk
  const int lm = lane & 15, kh = (lane >> 4) * 2, mh = (lane >> 4) * 8;
  const float* xn = xmod + (size_t)n * CCH * HW;
  const float* A  = W1 + (size_t)band * MID * CCH;

  v8f s_run[4] = {};
  v8f q_run[4] = {};
  for (int sb = 0; sb < 8; ++sb) {
    const int pix = chunk * 128 + sb * 16 + lm;
    v8f acc[4] = {};
    for (int k0 = 0; k0 < CCH; k0 += 4) {
      int kk = k0 + kh;
      v2f bb;
      bb.x = xn[(size_t)kk * HW + pix];
      bb.y = xn[(size_t)(kk + 1) * HW + pix];
#pragma unroll
      for (int t = 0; t < 4; ++t) {
        v2f a = *(const v2f*)(A + (t * 16 + lm) * CCH + kk);
        acc[t] = wmma4(a, bb, acc[t]);
      }
    }
#pragma unroll
    for (int t = 0; t < 4; ++t) {
      s_run[t] += acc[t];
      q_run[t] += acc[t] * acc[t];
    }
  }
  const int nb = n * NBAND + band;
#pragma unroll
  for (int t = 0; t < 4; ++t)
#pragma unroll
    for (int v = 0; v < 8; ++v) {
      float s = s_run[t][v], q = q_run[t][v];
#pragma unroll
      for (int off = 1; off < 16; off <<= 1) {      // reduce 16-lane halves
        s += __shfl_xor(s, off, 32);
        q += __shfl_xor(q, off, 32);
      }
      if (lm == 0) {                                // lanes 0 and 16 write
        int m = t * 16 + v + mh;
        psum[((size_t)nb * MID + m) * 128 + chunk] = s;
        psq [((size_t)nb * MID + m) * 128 + chunk] = q;
      }
    }
}

// ---------------------------------------------------------------------------
// 4) reduce partials -> mean, rsqrt(var + eps)
// ---------------------------------------------------------------------------
__global__ void k_reduce_stats(const float* __restrict__ psum,
                               const float* __restrict__ psq,
                               float* __restrict__ mean,
                               float* __restrict__ rstd) {
  int i = blockIdx.x * blockDim.x + threadIdx.x;
  if (i >= NBATCH * NBAND * MID) return;
  float s = 0.0f, q = 0.0f;
  for (int j = 0; j < 128; ++j) { s += psum[(size_t)i * 128 + j]; q += psq[(size_t)i * 128 + j]; }
  float mu  = s * (1.0f / (float)HW);
  float var = q * (1.0f / (float)HW) - mu * mu;
  mean[i] = mu;
  rstd[i] = rsqrtf(fmaxf(var, 0.0f) + 1e-5f);
}

// ---------------------------------------------------------------------------
// 5) band chain. Block = (image, 128-pixel chunk); wave = band; 8 sub-tiles:
//    h1 (WMMA) -> IN+gelu -> LDS -> W2 (WMMA) -> gelu -> LDS -> W3 dot -> out
// ---------------------------------------------------------------------------
__global__ __launch_bounds__(256) void k_bands(
    const float* __restrict__ xmod, const float* __restrict__ W1,
    const float* __restrict__ W2, const float* __restrict__ W3,
    const float* __restrict__ mean, const float* __restrict__ rstd,
    float* __restrict__ out) {
  __shared__ float lds[8][MID * 17];
  const int band  = threadIdx.x >> 5;               // wave = band
  const int lane  = threadIdx.x & 31;
  const int n     = blockIdx.x >> 7;
  const int chunk = blockIdx.x & 127;
  const int lm = lane & 15, kh = (lane >> 4) * 2, mh = (lane >> 4) * 8;
  const int nb = n * NBAND + band;
  const float* xn = xmod + (size_t)n * CCH * HW;
  const float* A  = W1 + (size_t)band * MID * CCH;
  const float* A2 = W2 + (size_t)band * 32 * MID;
  const float* w3 = W3 + (size_t)band * 32;
  const float* mu = mean + (size_t)nb * MID;
  const float* rs = rstd + (size_t)nb * MID;
  float* sm = &lds[band][0];

  for (int sb = 0; sb < 8; ++sb) {
    const int pix = chunk * 128 + sb * 16 + lm;

    // h1 = W1[b] @ xmod[n]   ([64 x 128] x [128 x 16])
    v8f acc[4] = {};
    for (int k0 = 0; k0 < CCH; k0 += 4) {
      int kk = k0 + kh;
      v2f bb;
      bb.x = xn[(size_t)kk * HW + pix];
      bb.y = xn[(size_t)(kk + 1) * HW + pix];
#pragma unroll
      for (int t = 0; t < 4; ++t) {
        v2f a = *(const v2f*)(A + (t * 16 + lm) * CCH + kk);
        acc[t] = wmma4(a, bb, acc[t]);
      }
    }
    // instance-norm + gelu -> per-wave LDS tile [64 x 16]
#pragma unroll
    for (int t = 0; t < 4; ++t)
#pragma unroll
      for (int v = 0; v < 8; ++v) {
        int m = t * 16 + v + mh;
        sm[m * 17 + lm] = gelu_f((acc[t][v] - mu[m]) * rs[m]);
      }

    // h2 = gelu(W2[b] @ h1n)   ([32 x 64] x [64 x 16])
    v8f acc2[2] = {};
    for (int k0 = 0; k0 < MID; k0 += 4) {
      int kk = k0 + kh;
      v2f bb;
      bb.x = sm[kk * 17 + lm];
      bb.y = sm[(kk + 1) * 17 + lm];
#pragma unroll
      for (int t = 0; t < 2; ++t) {
        v2f a = *(const v2f*)(A2 + (t * 16 + lm) * MID + kk);
        acc2[t] = wmma4(a, bb, acc2[t]);
      }
    }
    // overwrite rows 0..31 with gelu(h2) (same-wave LDS is in-order)
#pragma unroll
    for (int t = 0; t < 2; ++t)
#pragma unroll
      for (int v = 0; v < 8; ++v)
        sm[(t * 16 + v + mh) * 17 + lm] = gelu_f(acc2[t][v]);

    // h3 = clip(gelu(W3[b] . h2), 0, 1): 32-tap dot per pixel (lanes 0..15)
    if (lane < 16) {
      float s = 0.0f;
#pragma unroll
      for (int k = 0; k < 32; ++k) s += w3[k] * sm[k * 17 + lane];
      out[(size_t)nb * HW + chunk * 128 + sb * 16 + lane] = clamp01(gelu_f(s));
    }
  }
}

// ---------------------------------------------------------------------------
// launch
// ---------------------------------------------------------------------------
extern "C" void kernel_launch(void* const* d_in, const int* in_sizes, int n_in,
                              void* d_out, int out_size, void* d_ws, size_t ws_size,
                              hipStream_t stream) {
  const float* x   = (const float*)d_in[0];
  const float* Wp  = (const float*)d_in[1];
  const float* Wm1 = (const float*)d_in[2];
  const float* Wm2 = (const float*)d_in[3];
  const float* W1  = (const float*)d_in[4];
  const float* W2  = (const float*)d_in[5];
  const float* W3  = (const float*)d_in[6];
  float* out = (float*)d_out;

  float* ws   = (float*)d_ws;
  float* emb  = ws;                                   // 32*16384      = 2 MB
  float* xmod = emb + (size_t)PDIM * HW;              // 8*128*16384   = 64 MB
  float* psum = xmod + (size_t)NBATCH * CCH * HW;     // 64*64*128     = 2 MB
  float* psq  = psum + (size_t)NBATCH * NBAND * MID * 128;
  float* mean = psq + (size_t)NBATCH * NBAND * MID * 128;   // 4096
  float* rstd = mean + (size_t)NBATCH * NBAND * MID;        // 4096
  (void)in_sizes; (void)n_in; (void)out_size; (void)ws_size;

  k_emb<<<HW / 256, 256, 0, stream>>>(Wp, emb);
  k_modulate<<<NBATCH * 128, 256, 0, stream>>>(x, emb, Wm1, Wm2, xmod);
  k_band_stats<<<NBATCH * 128, 256, 0, stream>>>(xmod, W1, psum, psq);
  k_reduce_stats<<<(NBATCH * NBAND * MID + 255) / 256, 256, 0, stream>>>(psum, psq, mean, rstd);
  k_bands<<<NBATCH * 128, 256, 0, stream>>>(xmod, W1, W2, W3, mean, rstd, out);
}